// SS2D_local_12000138625311
// MI455X (gfx1250) — compile-verified
//
#include <hip/hip_runtime.h>
#include <hip/hip_bf16.h>

// ---------------- problem constants ----------------
#define BB      2
#define HH      64
#define WW      64
#define LL      4096            // H*W
#define DMODEL  128
#define DINNER  256
#define NSTATE  16
#define DTRANK  8
#define NPROJ   40              // DTRANK + 2*NSTATE
#define NDIR    4

typedef __attribute__((ext_vector_type(2))) float v2f;
typedef __attribute__((ext_vector_type(8))) float v8f;

// =====================================================================
// Async global->LDS helpers (gfx1250 ASYNCcnt path).
// LDS operand is the byte offset within the workgroup's LDS allocation;
// per the ISA flat->LDS mapping the low 32 bits of a generic shared
// pointer ARE that offset.
// =====================================================================
__device__ __forceinline__ void async_ld16(unsigned ldsOff, const float* g) {
  asm volatile("global_load_async_to_lds_b128 %0, %1, off"
               :: "v"(ldsOff), "v"(g) : "memory");
}
#define WAIT_ASYNC_LE2() asm volatile("s_wait_asynccnt 0x2" ::: "memory")
#define WAIT_ASYNC_0()   asm volatile("s_wait_asynccnt 0x0" ::: "memory")

// =====================================================================
// Generic fp32 WMMA GEMM:  C[M,N] = A[M,K] * W[N,K]^T
// Block = 128 threads (4 waves); block tile 64(M) x NT*16(N).
// Wave w owns rows [16w,16w+16) and all NT column tiles: the A fragment
// is loaded once per k-step and reused across NT WMMAs (4x density).
// K multiple of 64, M multiple of 64.
// =====================================================================
template <int NT>
__global__ __launch_bounds__(128)
void gemm_wmma_f32(const float* __restrict__ A, const float* __restrict__ W,
                   float* __restrict__ C, int K, int lda, int ldw, int ldc) {
  __shared__ __align__(16) float sA[64][68];
  __shared__ __align__(16) float sW[NT * 16][68];
  const int tid   = threadIdx.x;
  const int lane  = tid & 31;
  const int wave  = tid >> 5;
  const int m0    = blockIdx.x * 64;
  const int nblk  = blockIdx.y * (NT * 16);
  const int mrow  = lane & 15;          // fragment row
  const int khalf = (lane >> 4) << 1;   // lanes 16-31 carry K=2,3

  const v8f vzero = {0.f, 0.f, 0.f, 0.f, 0.f, 0.f, 0.f, 0.f};
  v8f acc[NT];
#pragma unroll
  for (int t = 0; t < NT; ++t) acc[t] = vzero;

  for (int kc = 0; kc < K; kc += 64) {
    // stage A tile 64x64 as float4 (global_load_b128, coalesced)
    for (int i = tid; i < 64 * 16; i += 128) {
      int r = i >> 4, c4 = (i & 15) << 2;
      const float4 v = *(const float4*)&A[(size_t)(m0 + r) * lda + kc + c4];
      *(float4*)&sA[r][c4] = v;
    }
    // stage W tile (NT*16)x64 as float4
    for (int i = tid; i < NT * 16 * 16; i += 128) {
      int r = i >> 4, c4 = (i & 15) << 2;
      const float4 v = *(const float4*)&W[(size_t)(nblk + r) * ldw + kc + c4];
      *(float4*)&sW[r][c4] = v;
    }
    __syncthreads();

#pragma unroll
    for (int k4 = 0; k4 < 64; k4 += 4) {
      v2f a;                                   // A frag: reused NT times
      a.x = sA[(wave << 4) + mrow][k4 + khalf];
      a.y = sA[(wave << 4) + mrow][k4 + khalf + 1];
#pragma unroll
      for (int t = 0; t < NT; ++t) {
        v2f b;                                 // B frag 4x16 (B[k][n] = W[n][k])
        b.x = sW[(t << 4) + mrow][k4 + khalf];
        b.y = sW[(t << 4) + mrow][k4 + khalf + 1];
        acc[t] = __builtin_amdgcn_wmma_f32_16x16x4_f32(
            /*neg_a=*/false, a, /*neg_b=*/false, b,
            /*c_mod=*/(short)0, acc[t], /*reuse_a=*/false, /*reuse_b=*/false);
      }
    }
    __syncthreads();
  }

  // C/D layout: VGPR j -> M = j + 8*(lane>=16), N = lane&15
  const int chi  = lane >> 4;
  const int cn   = lane & 15;
  const int crow = m0 + (wave << 4) + (chi << 3);
#pragma unroll
  for (int t = 0; t < NT; ++t)
#pragma unroll
    for (int j = 0; j < 8; ++j)
      C[(size_t)(crow + j) * ldc + nblk + (t << 4) + cn] = acc[t][j];
}

// =====================================================================
// Depthwise 3x3 conv (pad 1) + bias + SiLU.
// =====================================================================
__global__ __launch_bounds__(256)
void conv_silu_kernel(const float* __restrict__ xz, const float* __restrict__ cw,
                      const float* __restrict__ cb, float* __restrict__ out) {
  const int d   = threadIdx.x;
  const int idx = blockIdx.x;           // b*L + l (row-major l)
  const int b   = idx >> 12;
  const int l   = idx & (LL - 1);
  const int h   = l >> 6, w = l & 63;
  float acc = cb[d];
#pragma unroll
  for (int dh = -1; dh <= 1; ++dh) {
    int hh = h + dh;
    if ((unsigned)hh >= HH) continue;
#pragma unroll
    for (int dw = -1; dw <= 1; ++dw) {
      int ww2 = w + dw;
      if ((unsigned)ww2 >= WW) continue;
      acc += cw[d * 9 + (dh + 1) * 3 + (dw + 1)] *
             xz[(size_t)((b << 12) + (hh << 6) + ww2) * 512 + d];
    }
  }
  out[(size_t)idx * DINNER + d] = acc / (1.f + __expf(-acc));   // SiLU
}

// =====================================================================
// Fold dt_proj into x_proj:  dW[k][d][j] = sum_r dtw[k,d,r]*xpw[k,r,j]
// =====================================================================
__global__ __launch_bounds__(256)
void fold_dtw_kernel(const float* __restrict__ dtw, const float* __restrict__ xpw,
                     float* __restrict__ dW) {
  int i = blockIdx.x * 256 + threadIdx.x;   // < 4*256*256
  int j = i & 255;
  int d = (i >> 8) & 255;
  int k = i >> 16;
  float s = 0.f;
#pragma unroll
  for (int r = 0; r < DTRANK; ++r)
    s += dtw[(k * 256 + d) * DTRANK + r] * xpw[(k * NPROJ + r) * 256 + j];
  dW[i] = s;
}

// =====================================================================
// Gather the 4 direction orderings: xs[z=b*4+k][lp][d] = xcs[b][h][w][d]
// =====================================================================
__global__ __launch_bounds__(256)
void gather_kernel(const float* __restrict__ xcs, float* __restrict__ xs) {
  const int d  = threadIdx.x;
  const int zi = blockIdx.x;              // z*L + lp
  const int z  = zi >> 12;
  const int lp = zi & (LL - 1);
  const int b  = z >> 2, k = z & 3;
  const int lq = (k >= 2) ? (LL - 1 - lp) : lp;
  int h, w;
  if ((k & 1) == 0) {                     // row-major within 16x16 window
    int hg = lq >> 10, wg = (lq >> 8) & 3, hi = (lq >> 4) & 15, wi = lq & 15;
    h = hg * 16 + hi; w = wg * 16 + wi;
  } else {                                // column-first
    int wg = lq >> 10, hg = (lq >> 8) & 3, wi = (lq >> 4) & 15, hi = lq & 15;
    h = hg * 16 + hi; w = wg * 16 + wi;
  }
  xs[(size_t)zi * DINNER + d] =
      xcs[(size_t)((b << 12) + (h << 6) + w) * DINNER + d];
}

// =====================================================================
// Selective scan. 8 blocks (b,k) x 256 threads (one channel each).
// 16 states/thread in registers.  Shared per-(b,k) B/C vectors are
// double-buffered in LDS with async global->LDS copies (ASYNCcnt):
// the next 8 KB chunk streams in while the current chunk is consumed.
// =====================================================================
#define SCHUNK 64
__global__ __launch_bounds__(256)
void scan_kernel(const float* __restrict__ xs, const float* __restrict__ dlt,
                 const float* __restrict__ bc, const float* __restrict__ A_logs,
                 const float* __restrict__ Ds, const float* __restrict__ dtb,
                 float* __restrict__ ys) {
  const int d  = threadIdx.x;
  const int z  = blockIdx.x;             // b*4 + k
  const int k  = z & 3;
  const int ch = k * 256 + d;
  float Arow[NSTATE], hst[NSTATE];
#pragma unroll
  for (int n = 0; n < NSTATE; ++n) {
    Arow[n] = -__expf(A_logs[ch * NSTATE + n]);
    hst[n]  = 0.f;
  }
  const float Dv   = Ds[ch];
  const float bias = dtb[ch];

  // interleaved [l][32] chunk (B in cols 0..15, C in 16..31), 2 buffers
  __shared__ __align__(16) float sBC[2][SCHUNK][32];
  const int zbase = z * LL;

  // per-thread async portion: 2 x float4 per chunk
  const int e0 = threadIdx.x;            // element = row*8 + quad
  const int e1 = threadIdx.x + 256;
  const unsigned base0 = (unsigned)(size_t)&sBC[0][0][0];
  const unsigned base1 = (unsigned)(size_t)&sBC[1][0][0];

  // prologue: stream chunk 0 into buffer 0
  {
    const float* src = bc + (size_t)zbase * 32;
    async_ld16(base0 + (e0 << 4), src + (e0 << 2));
    async_ld16(base0 + (e1 << 4), src + (e1 << 2));
  }

  const int nchunks = LL / SCHUNK;
  for (int ci = 0; ci < nchunks; ++ci) {
    const int buf = ci & 1;
    if (ci + 1 < nchunks) {
      // stream next chunk into the other buffer, then drain this chunk's
      // 2 ops (in-order completion => wait <=2 retires the older pair)
      const unsigned nb = (buf ? base0 : base1);
      const float* src = bc + (size_t)(zbase + (ci + 1) * SCHUNK) * 32;
      async_ld16(nb + (e0 << 4), src + (e0 << 2));
      async_ld16(nb + (e1 << 4), src + (e1 << 2));
      WAIT_ASYNC_LE2();
    } else {
      WAIT_ASYNC_0();
    }
    __syncthreads();

    if (ci + 1 < nchunks)                 // pull per-thread u toward L2
      __builtin_prefetch(&xs[(size_t)(zbase + (ci + 1) * SCHUNK) * DINNER + d], 0, 1);

    const int lc = ci * SCHUNK;
    for (int li = 0; li < SCHUNK; ++li) {
      const int l = lc + li;
      float u    = xs[(size_t)(zbase + l) * DINNER + d];
      float draw = dlt[(size_t)(zbase + l) * DINNER + d] + bias;
      float dt   = (draw > 20.f) ? draw : __logf(1.f + __expf(draw)); // softplus
      float du   = dt * u;
      float y    = 0.f;
#pragma unroll
      for (int n = 0; n < NSTATE; ++n) {
        float dA = __expf(dt * Arow[n]);
        hst[n] = fmaf(dA, hst[n], du * sBC[buf][li][n]);
        y      = fmaf(hst[n], sBC[buf][li][NSTATE + n], y);
      }
      ys[(size_t)(zbase + l) * DINNER + d] = y + u * Dv;
    }
    __syncthreads();                      // all readers done before re-fill
  }
}

// =====================================================================
// Merge 4 directions + LayerNorm(256) + SiLU(z) gate.
// =====================================================================
__global__ __launch_bounds__(256)
void merge_kernel(const float* __restrict__ ys, const float* __restrict__ xz,
                  const float* __restrict__ lnw, const float* __restrict__ lnb,
                  float* __restrict__ g) {
  __shared__ float red[256];
  const int d  = threadIdx.x;
  const int bl = blockIdx.x;             // b*L + l
  const int b  = bl >> 12;
  const int l  = bl & (LL - 1);
  const int t  = (l & 63) * 64 + (l >> 6);     // (l%W)*H + l/W
  const int base = b * NDIR * LL;

  float v = ys[(size_t)(base + l) * DINNER + d]
          + ys[(size_t)(base + 1 * LL + t) * DINNER + d]
          + ys[(size_t)(base + 2 * LL + (LL - 1 - l)) * DINNER + d]
          + ys[(size_t)(base + 3 * LL + (LL - 1 - t)) * DINNER + d];

  red[d] = v; __syncthreads();
  for (int s = 128; s > 0; s >>= 1) { if (d < s) red[d] += red[d + s]; __syncthreads(); }
  const float mu = red[0] * (1.f / 256.f);
  __syncthreads();
  const float dv = v - mu;
  red[d] = dv * dv; __syncthreads();
  for (int s = 128; s > 0; s >>= 1) { if (d < s) red[d] += red[d + s]; __syncthreads(); }
  const float var = red[0] * (1.f / 256.f);

  float yn = dv * rsqrtf(var + 1e-5f) * lnw[d] + lnb[d];
  float zv = xz[(size_t)bl * 512 + 256 + d];
  g[(size_t)bl * DINNER + d] = yn * (zv / (1.f + __expf(-zv)));
}

// =====================================================================
// host-side orchestration
// =====================================================================
extern "C" void kernel_launch(void* const* d_in, const int* in_sizes, int n_in,
                              void* d_out, int out_size, void* d_ws, size_t ws_size,
                              hipStream_t stream) {
  const float* x     = (const float*)d_in[0];
  const float* W_in  = (const float*)d_in[1];
  const float* convw = (const float*)d_in[2];
  const float* convb = (const float*)d_in[3];
  const float* xpw   = (const float*)d_in[4];
  const float* dtw   = (const float*)d_in[5];
  const float* dtb   = (const float*)d_in[6];
  const float* Alog  = (const float*)d_in[7];
  const float* Dsp   = (const float*)d_in[8];
  const float* lnw   = (const float*)d_in[9];
  const float* lnb   = (const float*)d_in[10];
  const float* Wout  = (const float*)d_in[11];
  float* out = (float*)d_out;

  // workspace layout (bytes)
  char* ws = (char*)d_ws;
  float* xz  = (float*)(ws);                                   // B*L*512   = 16 MB
  float* xcs = (float*)(ws + (size_t)16777216);                // B*L*256   =  8 MB
  float* dW  = (float*)(ws + (size_t)25165824);                // 4*256*256 =  1 MB
  float* xs  = (float*)(ws + (size_t)26214400);                // 8*L*256   = 32 MB
  float* dlt = (float*)(ws + (size_t)59768832);                // 8*L*256   = 32 MB
  float* bc  = (float*)(ws + (size_t)93323264);                // 8*L*32    =  4 MB
  float* ysb = (float*)(ws + (size_t)97517568);                // 8*L*256   = 32 MB
  float* g   = xcs;                                            // reuse (dead after gather)

  const int M = BB * LL;                                       // 8192 rows

  // 1) in-projection: xz = x @ W_in^T   (8192 x 128 x 512)
  gemm_wmma_f32<4><<<dim3(M / 64, 512 / 64), 128, 0, stream>>>(
      x, W_in, xz, DMODEL, DMODEL, DMODEL, 512);

  // 2) depthwise conv + SiLU
  conv_silu_kernel<<<M, 256, 0, stream>>>(xz, convw, convb, xcs);

  // 3) fold dt_proj into x_proj
  fold_dtw_kernel<<<(NDIR * 256 * 256) / 256, 256, 0, stream>>>(dtw, xpw, dW);

  // 4) directional gathers
  gather_kernel<<<BB * NDIR * LL, 256, 0, stream>>>(xcs, xs);

  // 5) per-(b,k) WMMA GEMMs: delta (4096x256x256) and B/C (4096x256x32)
  for (int z = 0; z < BB * NDIR; ++z) {
    const int kk = z & 3;
    const float* Az = xs + (size_t)z * LL * DINNER;
    gemm_wmma_f32<4><<<dim3(LL / 64, 256 / 64), 128, 0, stream>>>(
        Az, dW + (size_t)kk * 256 * 256, dlt + (size_t)z * LL * DINNER,
        DINNER, DINNER, DINNER, DINNER);
    gemm_wmma_f32<2><<<dim3(LL / 64, 1), 128, 0, stream>>>(
        Az, xpw + (size_t)kk * NPROJ * DINNER + (size_t)DTRANK * DINNER,
        bc + (size_t)z * LL * 32, DINNER, DINNER, DINNER, 32);
  }

  // 6) selective scan (async double-buffered B/C)
  scan_kernel<<<BB * NDIR, 256, 0, stream>>>(xs, dlt, bc, Alog, Dsp, dtb, ysb);

  // 7) merge + LN + gate
  merge_kernel<<<M, 256, 0, stream>>>(ysb, xz, lnw, lnb, g);

  // 8) out-projection: out = g @ W_out^T  (8192 x 256 x 128)
  gemm_wmma_f32<4><<<dim3(M / 64, 128 / 64), 128, 0, stream>>>(
      g, Wout, out, DINNER, DINNER, DINNER, DMODEL);
}